// MyLoss_11055245820044
// MI455X (gfx1250) — compile-verified
//
#include <hip/hip_runtime.h>
#include <math.h>

// Problem constants (reference: inputs [4096, 32000] f32, targets [4096] int)
#define B_ROWS 4096
#define C_COLS 32000
#define NV4    (C_COLS / 4)                 // 8000 float4 per row
#define TPB    256                          // 8 wave32 per block
#define NWAVES (TPB / 32)
#define NSTAGE ((NV4 + TPB - 1) / TPB)      // 32 stages
#define TAIL_LANES (NV4 - (NSTAGE - 1) * TPB)  // 64 valid lanes in final stage
#define QD     4                            // async pipeline depth (16 KB LDS)

// ---- CDNA5 async global->LDS helpers (ASYNCcnt-tracked DMA-style copies) ----

// Raw 32-bit LDS byte address of a __shared__ object (addrspace(3) ptrs are 32-bit).
__device__ __forceinline__ unsigned lds_addr_of(const void* p) {
    return (unsigned)(unsigned long long)
        (__attribute__((address_space(3))) const void*)p;
}

// global_load_async_to_lds_b128: VDST = per-lane LDS byte address,
// VADDR = per-lane 32-bit byte offset, SADDR = 64-bit scalar base (GVS mode).
__device__ __forceinline__ void async_load_f4_to_lds(unsigned lds_byte_addr,
                                                     unsigned voffset_bytes,
                                                     unsigned long long sbase) {
    asm volatile("global_load_async_to_lds_b128 %0, %1, %2"
                 :
                 : "v"(lds_byte_addr), "v"(voffset_bytes), "s"(sbase)
                 : "memory");
}

__device__ __forceinline__ void async_wait_le(int n) {  // n is compile-time const
    if      (n >= 3) asm volatile("s_wait_asynccnt 3" ::: "memory");
    else if (n == 2) asm volatile("s_wait_asynccnt 2" ::: "memory");
    else if (n == 1) asm volatile("s_wait_asynccnt 1" ::: "memory");
    else             asm volatile("s_wait_asynccnt 0" ::: "memory");
}

// Online-softmax pair merge: (m,s) <- merge((m,s), (mo,so))
__device__ __forceinline__ void osm_merge(float& m, float& s, float mo, float so) {
    float mn = fmaxf(m, mo);
    s = s * __expf(m - mn) + so * __expf(mo - mn);
    m = mn;
}

// -----------------------------------------------------------------------------
// Kernel 1: per-row online softmax statistics -> per-row loss terms.
// Identity: for softmax p vs one-hot(t),  sum|p - onehot| = 2*(1 - p_t),
// so d = 2*(1 - p_t)/C; loss1_row = exp(d + d^2); ce_row = logZ - x_t.
// -----------------------------------------------------------------------------
__global__ __launch_bounds__(TPB) void row_stats_kernel(
    const float* __restrict__ inputs,
    const int*   __restrict__ targets,
    float*       __restrict__ ws_l1,
    float*       __restrict__ ws_ce) {

    __shared__ float4 stage[QD][TPB];    // 16 KB quad buffer
    __shared__ float  red_m[NWAVES];
    __shared__ float  red_s[NWAVES];

    const int row = blockIdx.x;
    const int tid = threadIdx.x;
    const unsigned long long base =
        (unsigned long long)(inputs + (size_t)row * C_COLS);

    unsigned ldsb[QD];
    #pragma unroll
    for (int b = 0; b < QD; ++b) ldsb[b] = lds_addr_of(&stage[b][tid]);

    // Prologue: fill the pipeline (stages 0..QD-1 are fully in-bounds).
    #pragma unroll
    for (int s = 0; s < QD; ++s)
        async_load_f4_to_lds(ldsb[s], (unsigned)(s * TPB + tid) * 16u, base);

    // Two independent online-softmax accumulators (even/odd stages) for ILP.
    float m0 = -INFINITY, s0 = 0.0f;
    float m1 = -INFINITY, s1 = 0.0f;

    #pragma unroll
    for (int s = 0; s < NSTAGE; ++s) {
        // Async loads complete in order per wave: retire stage s.
        async_wait_le(NSTAGE - 1 - s < QD - 1 ? NSTAGE - 1 - s : QD - 1);

        const bool tail = (s == NSTAGE - 1);
        if (!tail || tid < TAIL_LANES) {
            float4 x = stage[s & (QD - 1)][tid];
            float m4 = fmaxf(fmaxf(x.x, x.y), fmaxf(x.z, x.w));
            float& m = (s & 1) ? m1 : m0;
            float& ss = (s & 1) ? s1 : s0;
            float mn = fmaxf(m, m4);
            float e  = __expf(x.x - mn) + __expf(x.y - mn) +
                       __expf(x.z - mn) + __expf(x.w - mn);
            ss = ss * __expf(m - mn) + e;
            m  = mn;
        }

        // Refill the buffer just consumed. Every lane issues (tail lanes clamp
        // their address) so ASYNCcnt stays uniform per wave.
        if (s + QD < NSTAGE) {
            unsigned i2 = (unsigned)((s + QD) * TPB + tid);
            if (i2 >= NV4) i2 = NV4 - 1;
            async_load_f4_to_lds(ldsb[s & (QD - 1)], i2 * 16u, base);
        }
    }

    // Merge the two accumulators, then wave32 xor-butterfly reduction.
    osm_merge(m0, s0, m1, s1);
    #pragma unroll
    for (int off = 16; off > 0; off >>= 1) {
        float mo = __shfl_xor(m0, off, 32);
        float so = __shfl_xor(s0, off, 32);
        osm_merge(m0, s0, mo, so);
    }

    const int wave = tid >> 5;
    if ((tid & 31) == 0) { red_m[wave] = m0; red_s[wave] = s0; }
    __syncthreads();

    if (tid == 0) {
        float M = red_m[0], S = red_s[0];
        #pragma unroll
        for (int w = 1; w < NWAVES; ++w) osm_merge(M, S, red_m[w], red_s[w]);

        const int   t    = targets[row];
        const float xt   = inputs[(size_t)row * C_COLS + t];
        const float logZ = M + logf(S);
        const float pt   = __expf(xt - logZ);
        const float d    = 2.0f * (1.0f - pt) * (1.0f / (float)C_COLS);
        ws_l1[row] = __expf(d + d * d);
        ws_ce[row] = logZ - xt;                  // -log p_t
    }
}

// -----------------------------------------------------------------------------
// Kernel 2: deterministic final reduction over rows -> scalar output.
// out = mean(loss1) + mean(ce)
// -----------------------------------------------------------------------------
__global__ __launch_bounds__(256) void final_reduce_kernel(
    const float* __restrict__ ws_l1,
    const float* __restrict__ ws_ce,
    float*       __restrict__ out) {

    __shared__ float sm[8];
    const int tid = threadIdx.x;

    float acc = 0.0f;
    for (int i = tid; i < B_ROWS; i += 256)
        acc += ws_l1[i] + ws_ce[i];

    #pragma unroll
    for (int off = 16; off > 0; off >>= 1)
        acc += __shfl_xor(acc, off, 32);

    if ((tid & 31) == 0) sm[tid >> 5] = acc;
    __syncthreads();

    if (tid == 0) {
        float t = 0.0f;
        #pragma unroll
        for (int w = 0; w < 8; ++w) t += sm[w];
        out[0] = t * (1.0f / (float)B_ROWS);
    }
}

// -----------------------------------------------------------------------------
extern "C" void kernel_launch(void* const* d_in, const int* in_sizes, int n_in,
                              void* d_out, int out_size, void* d_ws, size_t ws_size,
                              hipStream_t stream) {
    (void)in_sizes; (void)n_in; (void)out_size; (void)ws_size;

    const float* inputs  = (const float*)d_in[0];
    const int*   targets = (const int*)d_in[1];

    float* ws_l1 = (float*)d_ws;            // [B_ROWS]
    float* ws_ce = ws_l1 + B_ROWS;          // [B_ROWS]

    row_stats_kernel<<<B_ROWS, TPB, 0, stream>>>(inputs, targets, ws_l1, ws_ce);
    final_reduce_kernel<<<1, 256, 0, stream>>>(ws_l1, ws_ce, (float*)d_out);
}